// _MatrixDecompositionBase_3693671875223
// MI455X (gfx1250) — compile-verified
//
#include <hip/hip_runtime.h>
#include <stdint.h>

// Problem constants (from reference: x (4,16,160,72,72) f32, bases (4,160,1) f32, S=R=1)
#define BATCH   4
#define DDIM    160
#define NDIM    82944           // 16*72*72
#define TN      64              // n-columns per block tile
#define NTILES  (NDIM / TN)     // 1296
#define EPSV    1e-6f
#define NSTEPS  4

typedef float v2f  __attribute__((ext_vector_type(2)));
typedef float v8f  __attribute__((ext_vector_type(8)));
typedef float f4v  __attribute__((ext_vector_type(4)));

// ---------------- CDNA5 async global->LDS copy helpers ----------------
__device__ __forceinline__ void async_copy_b128(uint32_t lds_byte, const float* gptr) {
  // GLOBAL_LOAD_ASYNC_TO_LDS_B128: copies 16 bytes from global memory to LDS at byte address vdst, tracked by ASYNCcnt
  asm volatile("global_load_async_to_lds_b128 %0, %1, off"
               :: "v"(lds_byte), "v"(gptr) : "memory");
}
__device__ __forceinline__ void wait_async0() {
#if __has_builtin(__builtin_amdgcn_s_wait_asynccnt)
  __builtin_amdgcn_s_wait_asynccnt(0);
#else
  asm volatile("s_wait_asynccnt 0" ::: "memory");
#endif
}

// ---------------- init: coef = 1 (softmax over R=1), bases <- input ----------------
__global__ void k_init(float* __restrict__ coef, float* __restrict__ bases,
                       const float* __restrict__ bases_in) {
  int i = blockIdx.x * blockDim.x + threadIdx.x;
  if (i < BATCH * NDIM) coef[i] = 1.0f;
  if (i < BATCH * DDIM) bases[i] = bases_in[i];
}

// ---------------- btb = ||b||^2 per batch ----------------
__global__ __launch_bounds__(DDIM) void k_btb(const float* __restrict__ bases,
                                              float* __restrict__ btb) {
  __shared__ float sh[DDIM];
  int bi = blockIdx.x, d = threadIdx.x;
  float v = bases[bi * DDIM + d];
  sh[d] = v * v;
  __syncthreads();
  if (d == 0) {
    float s = 0.f;
    for (int i = 0; i < DDIM; ++i) s += sh[i];
    btb[bi] = s;
  }
}

// ---------------- fused NMF step over one x pass ----------------
// Per block: stage X[bi][0:160][n0:n0+64] into LDS (async), compute
// nc[n] = sum_d X[d,n] b[d] with WMMA f32 16x16x4, update coef, then
// (if do_nb) produce num_b / ctc partials from the same LDS tile.
__global__ __launch_bounds__(128) void k_fused(
    const float* __restrict__ x, float* __restrict__ coef,
    const float* __restrict__ bases, const float* __restrict__ btb,
    float* __restrict__ nb_part, float* __restrict__ ctc_part, int do_nb)
{
  __shared__ float tile[DDIM * TN];   // 40 KB
  __shared__ float bsh[DDIM];
  __shared__ float ncbuf[TN];
  __shared__ float cbuf[TN];

  const int tid     = threadIdx.x;
  const int bi      = blockIdx.y;
  const int tileIdx = blockIdx.x;
  const int n0      = tileIdx * TN;
  const float* xb   = x + (size_t)bi * DDIM * NDIM;

  for (int d = tid; d < DDIM; d += 128) bsh[d] = bases[bi * DDIM + d];

  // ---- async tile load: 160 rows x 64 floats = 2560 float4, 20 per thread ----
  const uint32_t tbase = (uint32_t)(uintptr_t)&tile[0];  // low 32 bits = LDS byte offset
#pragma unroll
  for (int i = 0; i < (DDIM * TN / 4) / 128; ++i) {
    int f   = tid + i * 128;
    int row = f >> 4;       // 16 float4 per row
    int c4  = f & 15;
    const float* g = xb + (size_t)row * NDIM + n0 + c4 * 4;
    async_copy_b128(tbase + (uint32_t)(row * TN + c4 * 4) * 4u, g);
  }
  wait_async0();
  __syncthreads();

  // ---- phase 1: nc via WMMA. wave w owns local columns [w*16, w*16+16) ----
  const int wave  = tid >> 5;
  const int lane  = tid & 31;
  const int jbase = wave * 16;
  const int mlo   = lane & 15;
  const bool hiHalf = (lane >= 16);

#if __has_builtin(__builtin_amdgcn_wmma_f32_16x16x4_f32)
  v8f acc = {0.f, 0.f, 0.f, 0.f, 0.f, 0.f, 0.f, 0.f};
#pragma unroll 4
  for (int kc = 0; kc < DDIM / 4; ++kc) {
    // A (16x4): lanes 0-15 hold K=d0,d0+1; lanes 16-31 hold K=d0+2,d0+3 (M = lane%16)
    // B (4x16): b replicated across all 16 result columns -> D[m, *] = nc[n0+jbase+m]
    const int d0 = kc * 4;
    const int ra = hiHalf ? d0 + 2 : d0;
    v2f a, b;
    a.x = tile[ra * TN + jbase + mlo];
    a.y = tile[(ra + 1) * TN + jbase + mlo];
    b.x = bsh[ra];
    b.y = bsh[ra + 1];
    acc = __builtin_amdgcn_wmma_f32_16x16x4_f32(
        /*neg_a=*/false, a, /*neg_b=*/false, b,
        /*c_mod=*/(short)0, acc, /*reuse_a=*/false, /*reuse_b=*/false);
  }
  // D layout: vgpr r, lanes 0-15 -> M=r ; lanes 16-31 -> M=8+r. Column 0 suffices.
  if (lane == 0) {
#pragma unroll
    for (int r = 0; r < 8; ++r) ncbuf[jbase + r] = acc[r];
  } else if (lane == 16) {
#pragma unroll
    for (int r = 0; r < 8; ++r) ncbuf[jbase + 8 + r] = acc[r];
  }
#else
  // VALU fallback: 16 lanes per wave each do a 160-length dot product
  if (lane < 16) {
    float s = 0.f;
    for (int d = 0; d < DDIM; ++d) s += tile[d * TN + jbase + lane] * bsh[d];
    ncbuf[jbase + lane] = s;
  }
#endif
  __syncthreads();

  // ---- coef multiplicative update ----
  const float btbv = btb[bi];
  if (tid < TN) {
    const size_t ci = (size_t)bi * NDIM + n0 + tid;
    float cold = coef[ci];
    float cnew = cold * ncbuf[tid] / (cold * btbv + EPSV);
    coef[ci]  = cnew;
    cbuf[tid] = cnew;
  }
  __syncthreads();

  // ---- phase 2: num_b / ctc partials from the same LDS tile ----
  if (do_nb) {
    for (int d = tid; d < DDIM; d += 128) {
      float s = 0.f;
#pragma unroll 8
      for (int jj = 0; jj < TN; ++jj) {
        int j = (jj + tid) & (TN - 1);   // bank-conflict-free rotation
        s += tile[d * TN + j] * cbuf[j];
      }
      nb_part[((size_t)bi * NTILES + tileIdx) * DDIM + d] = s;
    }
    if (tid == 0) {
      float s = 0.f;
      for (int j = 0; j < TN; ++j) s += cbuf[j] * cbuf[j];
      ctc_part[bi * NTILES + tileIdx] = s;
    }
  }
}

// ---------------- bases multiplicative update + new btb ----------------
__global__ __launch_bounds__(DDIM) void k_bases(
    float* __restrict__ bases, float* __restrict__ btb,
    const float* __restrict__ nb_part, const float* __restrict__ ctc_part)
{
  __shared__ float sh[DDIM];
  __shared__ float shc;
  const int bi = blockIdx.x, d = threadIdx.x;
  if (d == 0) {
    float s = 0.f;
    for (int t = 0; t < NTILES; ++t) s += ctc_part[bi * NTILES + t];
    shc = s;
  }
  float nb = 0.f;
  for (int t = 0; t < NTILES; ++t)
    nb += nb_part[((size_t)bi * NTILES + t) * DDIM + d];
  __syncthreads();
  const float ctc = shc;
  const float bo  = bases[bi * DDIM + d];
  const float bn  = bo * nb / (bo * ctc + EPSV);
  bases[bi * DDIM + d] = bn;
  sh[d] = bn * bn;
  __syncthreads();
  if (d == 0) {
    float s = 0.f;
    for (int i = 0; i < DDIM; ++i) s += sh[i];
    btb[bi] = s;
  }
}

// ---------------- out[b,d,n] = bases[b,d] * coef[b,n] (rank-1), streamed NT ----------------
__global__ __launch_bounds__(256) void k_out(
    const float* __restrict__ coef, const float* __restrict__ bases,
    float* __restrict__ out)
{
  const int n4 = blockIdx.x * 256 + threadIdx.x;  // 0 .. N/4-1
  const int d  = blockIdx.y;
  const int bi = blockIdx.z;
  const f4v* c4 = (const f4v*)(coef + (size_t)bi * NDIM);
  f4v c = c4[n4];
  const float s = bases[bi * DDIM + d];
  f4v o = { s * c.x, s * c.y, s * c.z, s * c.w };
  f4v* op = (f4v*)(out + ((size_t)bi * DDIM + d) * NDIM);
  __builtin_nontemporal_store(o, &op[n4]);   // 212 MB pure stream, keep L2 for coef
}

extern "C" void kernel_launch(void* const* d_in, const int* in_sizes, int n_in,
                              void* d_out, int out_size, void* d_ws, size_t ws_size,
                              hipStream_t stream) {
  const float* x        = (const float*)d_in[0];   // (4,16,160,72,72) == flat [4][160][82944]
  const float* bases_in = (const float*)d_in[1];   // (4,160,1)
  float* out = (float*)d_out;

  // workspace layout (floats): coef | bases | btb | nb_part | ctc_part  (~4.7 MB)
  float* ws       = (float*)d_ws;
  float* coef     = ws;                                   // BATCH*NDIM
  float* bases    = coef + (size_t)BATCH * NDIM;          // BATCH*DDIM
  float* btb      = bases + BATCH * DDIM;                 // BATCH
  float* nb_part  = btb + BATCH;                          // BATCH*NTILES*DDIM
  float* ctc_part = nb_part + (size_t)BATCH * NTILES * DDIM; // BATCH*NTILES
  (void)in_sizes; (void)n_in; (void)out_size; (void)ws_size;

  k_init<<<(BATCH * NDIM + 255) / 256, 256, 0, stream>>>(coef, bases, bases_in);
  k_btb<<<BATCH, DDIM, 0, stream>>>(bases, btb);

  for (int s = 0; s < NSTEPS; ++s) {
    k_fused<<<dim3(NTILES, BATCH), 128, 0, stream>>>(x, coef, bases, btb,
                                                     nb_part, ctc_part, 1);
    k_bases<<<BATCH, DDIM, 0, stream>>>(bases, btb, nb_part, ctc_part);
  }
  // final _compute_coef (no bases update), then rank-1 output
  k_fused<<<dim3(NTILES, BATCH), 128, 0, stream>>>(x, coef, bases, btb,
                                                   nb_part, ctc_part, 0);
  k_out<<<dim3(NDIM / 4 / 256, DDIM, BATCH), 256, 0, stream>>>(coef, bases, out);
}